// GCN_one_layer_19602230739360
// MI455X (gfx1250) — compile-verified
//
#include <hip/hip_runtime.h>

typedef __attribute__((ext_vector_type(2))) float v2f;
typedef __attribute__((ext_vector_type(8))) float v8f;
typedef __attribute__((ext_vector_type(4))) int   v4i;

// Relaxed, agent-scope fp32 atomic add without return -> global_atomic_add_f32
__device__ __forceinline__ void atomic_add_f32(float* p, float v) {
  __hip_atomic_fetch_add(p, v, __ATOMIC_RELAXED, __HIP_MEMORY_SCOPE_AGENT);
}

// ---------------------------------------------------------------------------
// K1: deg[i] = 1 (self loop), out[i] = 0
// ---------------------------------------------------------------------------
__global__ __launch_bounds__(256) void k_init(float* __restrict__ deg,
                                              float* __restrict__ out, int N) {
  int i = blockIdx.x * blockDim.x + threadIdx.x;
  if (i < N) {
    deg[i] = 1.0f;
    out[i] = 0.0f;
  }
}

// ---------------------------------------------------------------------------
// K2: deg[col[e]] += 1 for all edges.
// Edge stream is 128 MB read-once -> NT loads (keep L2 for the hot deg array);
// atomics land in L2 (deg = 4 MB, fully resident).
// ---------------------------------------------------------------------------
__global__ __launch_bounds__(256) void k_degree(const int* __restrict__ col,
                                                float* __restrict__ deg, int E) {
  int tid = blockIdx.x * blockDim.x + threadIdx.x;
  int stride = gridDim.x * blockDim.x;
  int nv = E >> 2;
  const v4i* c4p = (const v4i*)col;
  for (int v = tid; v < nv; v += stride) {
    v4i c = __builtin_nontemporal_load(c4p + v);   // global_load_b128 th:NT
    atomic_add_f32(&deg[c.x], 1.0f);
    atomic_add_f32(&deg[c.y], 1.0f);
    atomic_add_f32(&deg[c.z], 1.0f);
    atomic_add_f32(&deg[c.w], 1.0f);
  }
  for (int e = (nv << 2) + tid; e < E; e += stride)
    atomic_add_f32(&deg[col[e]], 1.0f);
}

// ---------------------------------------------------------------------------
// K3: per 16-node tile (one wave): h = x @ W^T via V_WMMA_F32_16X16X4_F32,
//     dis = rsqrt(deg) (deg overwritten in place), g = dis*h, store h.
// A (16x4 f32, 2 VGPRs): lanes 0-15 hold {K0,K1}, lanes 16-31 hold {K2,K3}.
// B (4x16 f32, 2 VGPRs): broadcast w to all 16 columns ->
//     lanes 0-15: {w0,w1}; lanes 16-31: {w2,0}.
// D (16x16 f32, 8 VGPRs): lane<16: c[r] = h[r]; lane>=16: c[r] = h[r+8].
// ---------------------------------------------------------------------------
__global__ __launch_bounds__(256) void k_node_wmma(
    const float* __restrict__ x, const float* __restrict__ W,
    float* __restrict__ deg_dis, float* __restrict__ g,
    float* __restrict__ h, int N) {
  const int lane = threadIdx.x & 31;
  const int wave = threadIdx.x >> 5;
  const int wavesPerBlock = blockDim.x >> 5;
  const int tile = blockIdx.x * wavesPerBlock + wave;
  const int base = tile * 16;
  if (base >= N) return;  // wave-uniform exit; survivors have EXEC all-1s

  const int m = lane & 15;
  int node = base + m;
  int nodec = (node < N) ? node : (N - 1);  // clamp, no divergence

  // A operand: row m of x, K-padded to 4 (K3 = 0). x is read-once: NT loads.
  const int koff = (lane >> 4) * 2;  // 0 for K0/K1 half, 2 for K2/K3 half
  float ax = __builtin_nontemporal_load(x + 3 * nodec + koff);
  float x1 = __builtin_nontemporal_load(x + 3 * nodec + 1);
  float ay = (lane < 16) ? x1 : 0.0f;
  v2f a;  a.x = ax;  a.y = ay;

  // B operand: W broadcast across all 16 output columns
  float w0 = W[0], w1 = W[1], w2 = W[2];
  v2f b;
  b.x = (lane < 16) ? w0 : w2;
  b.y = (lane < 16) ? w1 : 0.0f;

  v8f c = {};
  c = __builtin_amdgcn_wmma_f32_16x16x4_f32(
      /*neg_a=*/false, a, /*neg_b=*/false, b,
      /*c_mod=*/(short)0, c, /*reuse_a=*/false, /*reuse_b=*/false);

  // Extract h[m]: lanes with (lane&15) < 8 each own one node.
  const int idx = lane & 15;
  float hv = c[0];
  hv = (idx == 1) ? c[1] : hv;
  hv = (idx == 2) ? c[2] : hv;
  hv = (idx == 3) ? c[3] : hv;
  hv = (idx == 4) ? c[4] : hv;
  hv = (idx == 5) ? c[5] : hv;
  hv = (idx == 6) ? c[6] : hv;
  hv = (idx == 7) ? c[7] : hv;

  if (idx < 8) {
    int n2 = base + (lane >> 4) * 8 + idx;
    if (n2 < N) {
      float d = deg_dis[n2];
      float r = rsqrtf(d);  // deg >= 1 always (self loop)
      deg_dis[n2] = r;
      g[n2] = r * hv;
      h[n2] = hv;
    }
  }
}

// ---------------------------------------------------------------------------
// K4: out[col[e]] += g[row[e]].  dis[col] is folded in at finalize, so each
//     edge needs one gather + one atomic; g and out (4 MB each) are
//     L2-resident, the 256 MB edge stream uses NT loads.
// ---------------------------------------------------------------------------
__global__ __launch_bounds__(256) void k_scatter(
    const int* __restrict__ row, const int* __restrict__ col,
    const float* __restrict__ g, float* __restrict__ out, int E) {
  int tid = blockIdx.x * blockDim.x + threadIdx.x;
  int stride = gridDim.x * blockDim.x;
  int nv = E >> 2;
  const v4i* r4p = (const v4i*)row;
  const v4i* c4p = (const v4i*)col;
  for (int v = tid; v < nv; v += stride) {
    v4i r = __builtin_nontemporal_load(r4p + v);   // global_load_b128 th:NT
    v4i c = __builtin_nontemporal_load(c4p + v);   // global_load_b128 th:NT
    atomic_add_f32(&out[c.x], g[r.x]);
    atomic_add_f32(&out[c.y], g[r.y]);
    atomic_add_f32(&out[c.z], g[r.z]);
    atomic_add_f32(&out[c.w], g[r.w]);
  }
  for (int e = (nv << 2) + tid; e < E; e += stride)
    atomic_add_f32(&out[col[e]], g[row[e]]);
}

// ---------------------------------------------------------------------------
// K5: out[i] = dis[i] * (edge_accum + dis[i]*h[i]) + b
// ---------------------------------------------------------------------------
__global__ __launch_bounds__(256) void k_finalize(
    float* __restrict__ out, const float* __restrict__ dis,
    const float* __restrict__ h, const float* __restrict__ b, int N) {
  int i = blockIdx.x * blockDim.x + threadIdx.x;
  if (i < N) {
    float r = dis[i];
    out[i] = r * (out[i] + r * h[i]) + b[0];
  }
}

extern "C" void kernel_launch(void* const* d_in, const int* in_sizes, int n_in,
                              void* d_out, int out_size, void* d_ws, size_t ws_size,
                              hipStream_t stream) {
  const float* x = (const float*)d_in[0];   // [N,3]
  const float* W = (const float*)d_in[1];   // [1,3]
  const float* b = (const float*)d_in[2];   // [1]
  const int*   ei = (const int*)d_in[3];    // [2,E] flat: rows then cols

  const int N = in_sizes[0] / 3;
  const int E = in_sizes[3] / 2;
  const int* row = ei;
  const int* col = ei + E;

  float* out = (float*)d_out;
  float* wsf = (float*)d_ws;
  float* deg_dis = wsf;                 // N floats: degree, then rsqrt(degree)
  float* g = wsf + (size_t)N;           // N floats: dis*h
  float* h = wsf + (size_t)2 * N;       // N floats: x @ W^T

  const int tpb = 256;
  const int nbN = (N + tpb - 1) / tpb;
  const int nbE = 4096;  // grid-stride over 32M edges; deep MLP for 23 TB/s

  k_init<<<nbN, tpb, 0, stream>>>(deg_dis, out, N);
  k_degree<<<nbE, tpb, 0, stream>>>(col, deg_dis, E);

  const int tiles = (N + 15) / 16;
  const int wavesPerBlock = tpb / 32;
  const int nbT = (tiles + wavesPerBlock - 1) / wavesPerBlock;
  k_node_wmma<<<nbT, tpb, 0, stream>>>(x, W, deg_dis, g, h, N);

  k_scatter<<<nbE, tpb, 0, stream>>>(row, col, g, out, E);
  k_finalize<<<nbN, tpb, 0, stream>>>(out, deg_dis, h, b, N);
}